// GaussianSplat_27582279975446
// MI455X (gfx1250) — compile-verified
//
#include <hip/hip_runtime.h>
#include <hip/hip_bf16.h>
#include <stdint.h>

typedef float v2f  __attribute__((ext_vector_type(2)));
typedef float v8f  __attribute__((ext_vector_type(8)));
typedef uint32_t u32x4 __attribute__((ext_vector_type(4)));
typedef int   i32x4 __attribute__((ext_vector_type(4)));
typedef int   i32x8 __attribute__((ext_vector_type(8)));

#define NMAX 2048

// DPP16 move: dst = row-permuted src; lanes with no valid source take `old`.
// CTRL: 0x110+n = row_shr:n, 0x150+n = row_share:n (broadcast lane n of row).
template <int CTRL>
__device__ __forceinline__ float dpp_movf(float src, float old) {
    return __int_as_float(__builtin_amdgcn_update_dpp(
        __float_as_int(old), __float_as_int(src), CTRL, 0xF, 0xF, false));
}

// ---------------------------------------------------------------------------
// Kernel 1: per-gaussian preprocessing.
// ws layout (floats): [0,N) depth ; [N,11N) unsorted 10-array SoA ;
//   [11N,21N) sorted SoA (k=0..5 poly coeffs G, 6 alpha(unused), 7..9 rgb) ;
//   [21N,25N) sorted rgb packed as float4.
// maha(p,n) expanded as degree-2 polynomial in pixel coords (6 coeffs), with
// -2*ln(alpha) folded into the constant term so alpha_map = exp(-0.5*maha').
// ---------------------------------------------------------------------------
__global__ void gs_preprocess(const float* __restrict__ pose,
                              const float* __restrict__ positions,
                              const float* __restrict__ scales,
                              const float* __restrict__ rotations,
                              const float* __restrict__ opacity,
                              const float* __restrict__ features,
                              float* __restrict__ ws, int N)
{
    int n = blockIdx.x * blockDim.x + threadIdx.x;
    if (n >= N) return;

    float px = positions[3*n+0], py = positions[3*n+1], pz = positions[3*n+2];
    float cam0 = pose[0]*px + pose[1]*py + pose[2]*pz  + pose[3];
    float cam1 = pose[4]*px + pose[5]*py + pose[6]*pz  + pose[7];
    float cam2 = pose[8]*px + pose[9]*py + pose[10]*pz + pose[11];
    float depth = cam2;
    float mx = cam0 / depth, my = cam1 / depth;

    float qw = rotations[4*n+0], qx = rotations[4*n+1],
          qy = rotations[4*n+2], qz = rotations[4*n+3];
    float qn = 1.0f / sqrtf(qw*qw + qx*qx + qy*qy + qz*qz);
    qw *= qn; qx *= qn; qy *= qn; qz *= qn;
    float R00 = 1.0f - 2.0f*(qy*qy + qz*qz);
    float R01 = 2.0f*(qx*qy - qw*qz);
    float R02 = 2.0f*(qx*qz + qw*qy);
    float R10 = 2.0f*(qx*qy + qw*qz);
    float R11 = 1.0f - 2.0f*(qx*qx + qz*qz);
    float R12 = 2.0f*(qy*qz - qw*qx);
    float R20 = 2.0f*(qx*qz - qw*qy);
    float R21 = 2.0f*(qy*qz + qw*qx);
    float R22 = 1.0f - 2.0f*(qx*qx + qy*qy);

    float sx = scales[3*n+0], sy = scales[3*n+1], sz = scales[3*n+2];
    float s0 = sx*sx, s1 = sy*sy, s2 = sz*sz;
    float C00 = R00*s0*R00 + R01*s1*R01 + R02*s2*R02;
    float C01 = R00*s0*R10 + R01*s1*R11 + R02*s2*R12;
    float C02 = R00*s0*R20 + R01*s1*R21 + R02*s2*R22;
    float C11 = R10*s0*R10 + R11*s1*R11 + R12*s2*R12;
    float C12 = R10*s0*R20 + R11*s1*R21 + R12*s2*R22;
    float C22 = R20*s0*R20 + R21*s1*R21 + R22*s2*R22;

    float fx = pose[0], fy = pose[5];
    float iz = 1.0f / pz;                      // reference uses world z here
    float J00 = fx*iz,  J02 = -fx*px*iz*iz;
    float J11 = fy*iz,  J12 = -fy*py*iz*iz;

    float v00 = J00*C00 + J02*C02;
    float v01 = J00*C01 + J02*C12;
    float v02 = J00*C02 + J02*C22;
    float v10 = J11*C01 + J12*C02;
    float v11 = J11*C11 + J12*C12;
    float v12 = J11*C12 + J12*C22;
    float a = v00*J00 + v02*J02;
    float b = v01*J11 + v02*J12;
    float c = v10*J00 + v12*J02;
    float d = v11*J11 + v12*J12;

    float idet = 1.0f / (a*d - b*c);
    float ia = d*idet, ib = -b*idet, ic = -c*idet, id_ = a*idet;
    float e  = ib + ic;

    float al = opacity[n];
    float G0 = ia;
    float G1 = e;
    float G2 = id_;
    float G3 = -(2.0f*ia*mx + e*my);
    float G4 = -(e*mx + 2.0f*id_*my);
    float G5 = ia*mx*mx + e*mx*my + id_*my*my
             - 2.0f * logf(al);                // fold alpha into the exponent

    float* depthA = ws;
    float* U = ws + N;
    depthA[n]   = depth;
    U[0*N + n]  = G0;  U[1*N + n] = G1;  U[2*N + n] = G2;
    U[3*N + n]  = G3;  U[4*N + n] = G4;  U[5*N + n] = G5;
    U[6*N + n]  = al;
    U[7*N + n]  = features[3*n+0];
    U[8*N + n]  = features[3*n+1];
    U[9*N + n]  = features[3*n+2];
}

// ---------------------------------------------------------------------------
// Kernel 2: stable rank of argsort(-depth) + scatter into sorted SoA + rgb4.
// ---------------------------------------------------------------------------
__global__ void gs_sort(float* __restrict__ ws, int N)
{
    int n = blockIdx.x * blockDim.x + threadIdx.x;
    if (n >= N) return;
    const float* depthA = ws;
    float di = depthA[n];
    int rank = 0;
    for (int j = 0; j < N; ++j) {
        float dj = depthA[j];
        rank += (dj > di) ? 1 : ((dj == di && j < n) ? 1 : 0);
    }
    const float* U = ws + N;
    float* S = ws + 11*N;
    #pragma unroll
    for (int k = 0; k < 10; ++k) S[k*N + rank] = U[k*N + n];
    float4* C4 = (float4*)(ws + 21*(size_t)N);
    C4[rank] = make_float4(U[7*N + n], U[8*N + n], U[9*N + n], 0.0f);
}

// ---------------------------------------------------------------------------
// Kernel 3: render. Each wave owns 16 consecutive pixels of one image row.
// Per 16-gaussian tile: two chained V_WMMA_F32_16X16X4_F32 produce maha for
// a 16x16 (pixel x gaussian) tile; alpha_map = exp(-0.5*maha'); the serial
// transmittance composite is a 16-wide DPP prefix product (row_shr 1/2/4/8),
// exclusive shift (row_shr:1) and lane-15 broadcast (row_share:15) — all
// VALU, no LDS traffic. G-coeff table is staged to LDS via the TDM.
// ---------------------------------------------------------------------------
__global__ __launch_bounds__(256) void gs_render(const float* __restrict__ ws,
                                                 float* __restrict__ out,
                                                 int N, int H, int W)
{
    __shared__ float ldsG[6 * NMAX];           // 48 KB: sorted poly coeffs
    const float* S = ws + 11*N;
    const float4* C4 = (const float4*)(ws + 21*(size_t)N);
    (void)S;

    // --- Stage G0..G5 (6*N contiguous floats) into LDS via Tensor Data Mover
    if (threadIdx.x < 32) {
        uint64_t gaddr  = (uint64_t)(uintptr_t)(ws + 11*(size_t)N);
        uint32_t ldsoff = (uint32_t)(uintptr_t)(void*)&ldsG[0];
        uint32_t total  = 6u * (uint32_t)N;    // f32 elements, fits 16 bits

        u32x4 g0;
        g0.x = 1u;                                           // count=1, user D#
        g0.y = ldsoff;                                       // lds_addr
        g0.z = (uint32_t)gaddr;                              // global_addr[31:0]
        g0.w = (uint32_t)((gaddr >> 32) & 0x1FFFFFFu)        // global_addr[56:32]
             | (2u << 30);                                   // type=2 ("image")

        i32x8 g1;
        g1[0] = (int)(2u << 16);                             // data_size=4B
        g1[1] = (int)((total & 0xFFFFu) << 16);              // tensor_dim0[15:0]
        g1[2] = (int)(((total >> 16) & 0xFFFFu) | (1u<<16)); // dim0 hi | dim1=1
        g1[3] = (int)((total & 0xFFFFu) << 16);              // tile_dim0 = total
        g1[4] = 0;                                           // tile_dim1/2 unused
        g1[5] = (int)total;                                  // tensor_dim0_stride
        g1[6] = 0;
        g1[7] = 0;

        i32x4 gz = {0, 0, 0, 0};
#if defined(__clang_major__) && (__clang_major__ >= 23)
        i32x8 gz8 = {0, 0, 0, 0, 0, 0, 0, 0};
        __builtin_amdgcn_tensor_load_to_lds(g0, g1, gz, gz, gz8, 0);
#else
        __builtin_amdgcn_tensor_load_to_lds(g0, g1, gz, gz, 0);
#endif
        __builtin_amdgcn_s_wait_tensorcnt(0);
    }
    __syncthreads();

    int wid  = (int)blockIdx.x * ((int)blockDim.x >> 5) + ((int)threadIdx.x >> 5);
    int lane = (int)threadIdx.x & 31;
    int m    = lane & 15;
    bool hi  = lane >= 16;

    int p0   = wid * 16;                       // first pixel of this wave
    int hrow = p0 / W;
    int wcol = (p0 % W) + m;
    float x = -1.0f + 2.0f * (float)wcol / (float)(W - 1);
    float y = -1.0f + 2.0f * (float)hrow / (float)(H - 1);

    // A-matrix (16x4 f32 layout): lanes 0-15 hold K0/K1, lanes 16-31 K2/K3.
    // WMMA#1: K = [x^2, xy, y^2, x]; WMMA#2: K = [y, 1, 0, 0]
    v2f A1, A2;
    if (!hi) { A1[0] = x*x; A1[1] = x*y; A2[0] = y;    A2[1] = 1.0f; }
    else     { A1[0] = y*y; A1[1] = x;   A2[0] = 0.0f; A2[1] = 0.0f; }

    float T[8], aR[8], aG[8], aB[8];
    #pragma unroll
    for (int r = 0; r < 8; ++r) { T[r] = 1.0f; aR[r] = aG[r] = aB[r] = 0.0f; }

    int k0 = hi ? 2 : 0;
    int NT = N >> 4;
    for (int t = 0; t < NT; ++t) {
        int n = (t << 4) + m;
        v2f B1, B2;
        B1[0] = ldsG[k0*N + n];
        B1[1] = ldsG[(k0+1)*N + n];
        B2[0] = hi ? 0.0f : ldsG[4*N + n];
        B2[1] = hi ? 0.0f : ldsG[5*N + n];

        v8f mh = {};
        mh = __builtin_amdgcn_wmma_f32_16x16x4_f32(false, A1, false, B1,
                                                   (short)0, mh, false, false);
        mh = __builtin_amdgcn_wmma_f32_16x16x4_f32(false, A2, false, B2,
                                                   (short)0, mh, false, false);

        float4 col = C4[n];

        // C/D layout: VGPR r = pixel row (r / r+8 by lane half), lane%16 =
        // gaussian column. Composite 16 gaussians in order with DPP.
        #pragma unroll
        for (int r = 0; r < 8; ++r) {
            float am = __expf(-0.5f * mh[r]);      // alpha folded into G5
            float u  = 1.0f - am;
            float incl = u;
            incl *= dpp_movf<0x111>(incl, 1.0f);   // row_shr:1
            incl *= dpp_movf<0x112>(incl, 1.0f);   // row_shr:2
            incl *= dpp_movf<0x114>(incl, 1.0f);   // row_shr:4
            incl *= dpp_movf<0x118>(incl, 1.0f);   // row_shr:8
            float excl = dpp_movf<0x111>(incl, 1.0f);
            float tot  = dpp_movf<0x15F>(incl, incl);  // row_share:15 bcast
            float wgt  = T[r] * excl * am;
            aR[r] += wgt * col.x;
            aG[r] += wgt * col.y;
            aB[r] += wgt * col.z;
            T[r] *= tot;
        }
    }

    // One-time reduction of each pixel row over the 16 gaussian-column lanes.
    int HW = H * W;
    #pragma unroll
    for (int r = 0; r < 8; ++r) {
        float vR = aR[r], vG = aG[r], vB = aB[r];
        #pragma unroll
        for (int s = 1; s < 16; s <<= 1) {
            vR += __shfl_xor(vR, s, 16);
            vG += __shfl_xor(vG, s, 16);
            vB += __shfl_xor(vB, s, 16);
        }
        if (m == 0) {
            int p = p0 + r + (hi ? 8 : 0);
            out[0*HW + p] = vR;
            out[1*HW + p] = vG;
            out[2*HW + p] = vB;
        }
    }
}

// ---------------------------------------------------------------------------
extern "C" void kernel_launch(void* const* d_in, const int* in_sizes, int n_in,
                              void* d_out, int out_size, void* d_ws, size_t ws_size,
                              hipStream_t stream) {
    const float* pose      = (const float*)d_in[0];
    const float* positions = (const float*)d_in[1];
    const float* scales    = (const float*)d_in[2];
    const float* rotations = (const float*)d_in[3];
    const float* opacity   = (const float*)d_in[4];
    const float* features  = (const float*)d_in[5];
    int N = in_sizes[1] / 3;          // 2048
    const int H = 96, W = 96;         // fixed by the reference

    float* ws  = (float*)d_ws;        // uses 25*N floats (~205 KB)
    float* out = (float*)d_out;       // 3*H*W f32

    int nb = (N + 255) / 256;
    gs_preprocess<<<nb, 256, 0, stream>>>(pose, positions, scales, rotations,
                                          opacity, features, ws, N);
    gs_sort<<<nb, 256, 0, stream>>>(ws, N);

    int pixel_blocks = (H * W) / 128; // 128 pixels (8 waves x 16) per block
    gs_render<<<pixel_blocks, 256, 0, stream>>>(ws, out, N, H, W);
}